// GHMCLoss_10737418240845
// MI455X (gfx1250) — compile-verified
//
#include <hip/hip_runtime.h>
#include <stdint.h>

#define BINS   10
#define BLOCK  256
#define WAVES  8          // BLOCK / 32 (wave32 on gfx1250)
#define STAGES 4          // async pipeline depth (power of two)

// Low 32 bits of a flat pointer into __shared__ == LDS byte address
// (ISA: LDS aperture maps addr[31:0] directly to the LDS offset).
__device__ __forceinline__ uint32_t lds_addr32(const void* p) {
    return (uint32_t)(uintptr_t)p;
}

// CDNA5 async copy: global -> LDS, 16 bytes per lane, tracked by ASYNCcnt.
__device__ __forceinline__ void async_copy_b128(uint32_t lds_off, const void* gptr) {
    asm volatile("global_load_async_to_lds_b128 %0, %1, off"
                 :: "v"(lds_off), "v"(gptr) : "memory");
}

__device__ __forceinline__ void ghm_accum(float p, float t,
                                          float* mySum, unsigned* myCnt) {
    float g = fabsf(p - t);                       // gradient norm
    int   u = (int)(g * 10.0f);                   // floor (g >= 0)
    u = (u > BINS - 1) ? (BINS - 1) : u;
    // t==1 -> log(p); t==0 -> log(1-p); both equal log(1-g) for binary t
    float val = fmaxf(__logf(1.0f - g), -100.0f);
    atomicAdd(mySum + u * 32, val);               // ds_add_f32, own slot -> deterministic
    atomicAdd(myCnt + u * 32, 1u);                // ds_add_u32
}

__global__ __launch_bounds__(BLOCK) void ghmc_pass1(
    const float* __restrict__ pred, const float* __restrict__ tgt,
    float* __restrict__ wsS, unsigned* __restrict__ wsC,
    long long nElem, long long nChunks, int nIter)
{
    __shared__ float4   bufP[WAVES][STAGES][32];   // 16 KB
    __shared__ float4   bufT[WAVES][STAGES][32];   // 16 KB
    __shared__ float    histS[WAVES][BINS][32];    // 10 KB  per-(wave,lane) sums
    __shared__ unsigned histC[WAVES][BINS][32];    // 10 KB  per-(wave,lane) counts
    __shared__ float    redS[BINS * 32];
    __shared__ unsigned redC[BINS * 32];

    const int tid = threadIdx.x;

    // zero histograms
    {
        float*    hs = &histS[0][0][0];
        unsigned* hc = &histC[0][0][0];
        for (int i = tid; i < WAVES * BINS * 32; i += BLOCK) { hs[i] = 0.0f; hc[i] = 0u; }
    }
    __syncthreads();

    const unsigned w    = (unsigned)tid >> 5;
    const unsigned lane = (unsigned)tid & 31u;

    float*    mySum = &histS[w][0][lane];          // stride 32 floats per bin
    unsigned* myCnt = &histC[w][0][lane];

    if (nChunks > 0) {
        // wave's first chunk (64-bit once; all loop math below is 32-bit)
        const long long gw = (long long)blockIdx.x * WAVES + w;
        const unsigned baseChunk = (unsigned)(gw * (long long)nIter * 32 + (long long)lane);
        const unsigned lastChunk = (unsigned)(nChunks - 1);

        const uint32_t ldsPbase = lds_addr32(&bufP[w][0][lane]);  // +512 per stage
        const uint32_t ldsTbase = lds_addr32(&bufT[w][0][lane]);

        unsigned issueChunk = baseChunk;           // chunk for next issued stage
        // ---- prologue: fill STAGES-1 buffers ----
        #pragma unroll
        for (int j = 0; j < STAGES - 1; ++j) {
            unsigned c = issueChunk < lastChunk ? issueChunk : lastChunk;  // v_min_u32
            async_copy_b128(ldsPbase + (uint32_t)j * 512u, pred + (size_t)c * 4);
            async_copy_b128(ldsTbase + (uint32_t)j * 512u, tgt  + (size_t)c * 4);
            issueChunk += 32u;
        }
        // ---- main pipelined loop ----
        unsigned procChunk = baseChunk;
        for (int i = 0; i < nIter; ++i) {
            {   // issue stage i+STAGES-1 (clamped; overwrites an already-consumed buffer)
                const uint32_t soff = (uint32_t)((i + STAGES - 1) & (STAGES - 1)) * 512u;
                unsigned c = issueChunk < lastChunk ? issueChunk : lastChunk;
                async_copy_b128(ldsPbase + soff, pred + (size_t)c * 4);
                async_copy_b128(ldsTbase + soff, tgt  + (size_t)c * 4);
                issueChunk += 32u;
            }
            // oldest pair (2 ops) must be complete: wait ASYNCcnt <= 2*(STAGES-1)
            asm volatile("s_wait_asynccnt %0" :: "i"(2 * (STAGES - 1)) : "memory");

            const int s = i & (STAGES - 1);
            float4 p4 = bufP[w][s][lane];          // ds_load_b128
            float4 t4 = bufT[w][s][lane];

            if (procChunk <= lastChunk) {
                ghm_accum(p4.x, t4.x, mySum, myCnt);
                ghm_accum(p4.y, t4.y, mySum, myCnt);
                ghm_accum(p4.z, t4.z, mySum, myCnt);
                ghm_accum(p4.w, t4.w, mySum, myCnt);
            }
            procChunk += 32u;
        }
        asm volatile("s_wait_asynccnt 0" ::: "memory");
    }

    // scalar tail (nElem % 4), done by one thread before the block reduction
    if (blockIdx.x == 0 && tid == 0) {
        for (long long e2 = nChunks * 4; e2 < nElem; ++e2) {
            ghm_accum(pred[e2], tgt[e2], &histS[0][0][0], &histC[0][0][0]);
        }
    }
    __syncthreads();

    // ---- block reduction (fixed order -> deterministic) ----
    for (int i = tid; i < BINS * 32; i += BLOCK) {
        float s = 0.0f; unsigned c = 0u;
        #pragma unroll
        for (int ww = 0; ww < WAVES; ++ww) {
            s += (&histS[ww][0][0])[i];
            c += (&histC[ww][0][0])[i];
        }
        redS[i] = s; redC[i] = c;
    }
    __syncthreads();
    if (tid < BINS) {
        float s = 0.0f; unsigned c = 0u;
        for (int l = 0; l < 32; ++l) { s += redS[tid * 32 + l]; c += redC[tid * 32 + l]; }
        wsS[(long long)blockIdx.x * BINS + tid] = s;
        wsC[(long long)blockIdx.x * BINS + tid] = c;
    }
}

__global__ __launch_bounds__(320) void ghmc_pass2(
    const float* __restrict__ wsS, const unsigned* __restrict__ wsC,
    float* __restrict__ out, int numBlocks)
{
    __shared__ float    fS[BINS][32];
    __shared__ unsigned fC[BINS][32];
    __shared__ float    binS[BINS];
    __shared__ unsigned binC[BINS];

    const int tid = threadIdx.x;
    const int b = tid >> 5, g = tid & 31;   // 10 waves: one bin per wave
    {
        float s = 0.0f; unsigned c = 0u;
        for (int k = g; k < numBlocks; k += 32) {   // fixed strided order
            s += wsS[(long long)k * BINS + b];
            c += wsC[(long long)k * BINS + b];
        }
        fS[b][g] = s; fC[b][g] = c;
    }
    __syncthreads();
    if (tid < BINS) {
        float s = 0.0f; unsigned c = 0u;
        for (int l = 0; l < 32; ++l) { s += fS[tid][l]; c += fC[tid][l]; }
        binS[tid] = s; binC[tid] = c;
    }
    __syncthreads();
    if (tid == 0) {
        int nn = 0; float acc = 0.0f;
        for (int bb = 0; bb < BINS; ++bb) {
            if (binC[bb] > 0u) nn++;
            float cnt = (binC[bb] > 0u) ? (float)binC[bb] : 1.0f;
            acc += binS[bb] / cnt;
        }
        out[0] = -acc / (float)((nn > 0) ? nn : 1);
    }
}

extern "C" void kernel_launch(void* const* d_in, const int* in_sizes, int n_in,
                              void* d_out, int out_size, void* d_ws, size_t ws_size,
                              hipStream_t stream)
{
    const float* pred = (const float*)d_in[0];
    const float* tgt  = (const float*)d_in[1];
    const long long nElem   = (long long)in_sizes[0];
    const long long nChunks = nElem / 4;                      // float4 chunks

    int numBlocks = 2048;
    const size_t bytesPerBlock = (size_t)BINS * (sizeof(float) + sizeof(unsigned));
    if ((size_t)numBlocks * bytesPerBlock > ws_size) {
        numBlocks = (int)(ws_size / bytesPerBlock);
        if (numBlocks < 1) numBlocks = 1;
    }

    long long totalTiles = (nChunks + 31) / 32;               // 32 float4 per wave-tile
    if (totalTiles < 1) totalTiles = 1;
    const long long wavesTotal = (long long)numBlocks * WAVES;
    int nIter = (int)((totalTiles + wavesTotal - 1) / wavesTotal);
    if (nIter < 1) nIter = 1;

    float*    wsS = (float*)d_ws;
    unsigned* wsC = (unsigned*)((char*)d_ws + (size_t)numBlocks * BINS * sizeof(float));

    ghmc_pass1<<<numBlocks, BLOCK, 0, stream>>>(pred, tgt, wsS, wsC, nElem, nChunks, nIter);
    ghmc_pass2<<<1, 320, 0, stream>>>(wsS, wsC, (float*)d_out, numBlocks);
}